// ShiftConv3d_v1_80900003988110
// MI455X (gfx1250) — compile-verified
//
#include <hip/hip_runtime.h>

typedef __attribute__((ext_vector_type(2))) float v2f;
typedef __attribute__((ext_vector_type(8))) float v8f;

#define B_    2
#define CIN   128
#define COUT  128
#define T_    18
#define H_    128
#define W_    128
#define KTOT  (CIN*9)        // 1152
#define CCHUNK 32            // cin channels staged per chunk
#define NCHUNK (CIN/CCHUNK)  // 4
#define KCH   (CCHUNK*9)     // 288 K-steps per chunk
#define NT    64             // spatial (w) tile per workgroup
#define KSTR  292            // Bt k-stride: 288 + 4 pad (bank-conflict-free: 292%64=36, gcd(36,64)=4)
#define RAWN  66             // raw w values per row (NT + 2 halo)
#define RAW_ELEMS (CCHUNK*3*RAWN)   // 6336

// -------------------- Kernel 1: implicit-GEMM 3x3 conv via WMMA f32 --------------------
// temp[b][co][t][h][w] = bias[co] + sum_{ci,kh,kw} w[co][ci][0][kh][kw] * x[b][ci][t][h+kh-1][w+kw-1]
__global__ __launch_bounds__(256) void conv3x3_wmma_f32(
    const float* __restrict__ x, const float* __restrict__ wgt,
    const float* __restrict__ bias, float* __restrict__ temp)
{
    // double-buffered im2col B panels, transposed: Bt[n][k] (k contiguous -> ds_load_b64 fragments)
    __shared__ __align__(16) float bt0[NT * KSTR];   // 74,752 B each
    __shared__ __align__(16) float bt1[NT * KSTR];

    const int tid  = threadIdx.x;
    const int lane = tid & 31;
    const int wave = tid >> 5;

    const int wbase = blockIdx.x * NT;        // w tile start
    const int h     = blockIdx.y;             // output row
    const int btz   = blockIdx.z;             // b*T + t
    const int b     = btz / T_;
    const int t     = btz - b * T_;

    const int mbase = wave * 16;              // this wave's Cout base
    const int m     = lane & 15;              // A row (M) / B-D column (N)
    const int kb    = (lane >> 4) * 2;        // K sub-offset per ISA A/B lane layout

    v8f acc0 = {}, acc1 = {}, acc2 = {}, acc3 = {};

    const size_t cstride = (size_t)T_ * H_ * W_;                          // per-channel stride
    const float* xslice  = x + ((size_t)(b * CIN) * T_ + t) * (H_ * W_);  // x[b][0][t]

    // ---- 1) pre-zero the OOB (padding) slots in BOTH buffers.
    // The OOB pattern over (n, k_local) is identical for every cin chunk, and async
    // loads below only ever write in-range slots, so these zeros persist.
    for (int idx = tid; idx < RAW_ELEMS; idx += 256) {
        int cl = idx / (3 * RAWN);
        int r  = idx - cl * (3 * RAWN);
        int rr = r / RAWN;
        int wl = r - rr * RAWN;
        int gw = wbase - 1 + wl;
        int gh = h - 1 + rr;
        if (!((unsigned)gw < (unsigned)W_ && (unsigned)gh < (unsigned)H_)) {
            const int kbase = cl * 9 + rr * 3;
            #pragma unroll
            for (int kw = 0; kw < 3; ++kw) {
                const int nn = wl - kw;
                if ((unsigned)nn < (unsigned)NT) {
                    bt0[nn * KSTR + kbase + kw] = 0.0f;
                    bt1[nn * KSTR + kbase + kw] = 0.0f;
                }
            }
        }
    }

    // ---- async im2col staging: each in-range x value -> <=3 async b32 copies into LDS slots
    auto stage_async = [&](int cc, float* buf) {
        for (int idx = tid; idx < RAW_ELEMS; idx += 256) {
            int cl = idx / (3 * RAWN);
            int r  = idx - cl * (3 * RAWN);
            int rr = r / RAWN;
            int wl = r - rr * RAWN;
            int gw = wbase - 1 + wl;
            int gh = h - 1 + rr;
            if ((unsigned)gw < (unsigned)W_ && (unsigned)gh < (unsigned)H_) {
                unsigned long long ga = (unsigned long long)(uintptr_t)
                    (xslice + (size_t)(cc + cl) * cstride + gh * W_ + gw);
                const int kbase = cl * 9 + rr * 3;
                #pragma unroll
                for (int kw = 0; kw < 3; ++kw) {
                    const int nn = wl - kw;
                    if ((unsigned)nn < (unsigned)NT) {
                        unsigned lo = (unsigned)(uintptr_t)(buf + nn * KSTR + kbase + kw);
                        asm volatile("global_load_async_to_lds_b32 %0, %1, off"
                                     :: "v"(lo), "v"(ga) : "memory");
                    }
                }
            }
        }
    };

    // prologue: stage chunk 0
    stage_async(0, bt0);
    asm volatile("s_wait_asynccnt 0x0" ::: "memory");
    __syncthreads();

    for (int ch = 0; ch < NCHUNK; ++ch) {
        // issue next chunk's async loads into the other buffer (overlaps WMMA below)
        if (ch + 1 < NCHUNK)
            stage_async((ch + 1) * CCHUNK, (ch & 1) ? bt0 : bt1);

        const float* cur = (ch & 1) ? bt1 : bt0;

        // ---- K loop: A = one global b64, B = one ds b64 per WMMA ----
        const float* wck = wgt + (size_t)(mbase + m) * KTOT + ch * KCH + kb;
        const float* btn = cur + m * KSTR + kb;   // n == m (lane&15)
        for (int ks = 0; ks < KCH; ks += 4) {
            v2f a  = *(const v2f*)(wck + ks);
            v2f b0 = *(const v2f*)(btn + ks);
            v2f b1 = *(const v2f*)(btn + 16 * KSTR + ks);
            v2f b2 = *(const v2f*)(btn + 32 * KSTR + ks);
            v2f b3 = *(const v2f*)(btn + 48 * KSTR + ks);
            acc0 = __builtin_amdgcn_wmma_f32_16x16x4_f32(false, a, false, b0, (short)0, acc0, false, false);
            acc1 = __builtin_amdgcn_wmma_f32_16x16x4_f32(false, a, false, b1, (short)0, acc1, false, false);
            acc2 = __builtin_amdgcn_wmma_f32_16x16x4_f32(false, a, false, b2, (short)0, acc2, false, false);
            acc3 = __builtin_amdgcn_wmma_f32_16x16x4_f32(false, a, false, b3, (short)0, acc3, false, false);
        }

        // my async issues must be complete before anyone reads the next buffer,
        // and everyone must be done reading `cur` before it is refilled
        asm volatile("s_wait_asynccnt 0x0" ::: "memory");
        __syncthreads();
    }

    // ---- store D (+bias): lane<16 -> M=r, lane>=16 -> M=8+r ; N = lane&15 ----
    float* tp = temp + ((size_t)(b * COUT) * T_ + t) * (H_ * W_);
    const int rsel = (lane >> 4) * 8;
    #pragma unroll
    for (int r = 0; r < 8; ++r) {
        const int cout = mbase + rsel + r;
        const float bv = bias[cout];
        const size_t o = (size_t)cout * cstride + (size_t)h * W_ + wbase + m;
        tp[o +  0] = acc0[r] + bv;
        tp[o + 16] = acc1[r] + bv;
        tp[o + 32] = acc2[r] + bv;
        tp[o + 48] = acc3[r] + bv;
    }
}

// -------------------- Kernel 2: temporal shift-add (edge-clamped) --------------------
__device__ __constant__ int kA0[11] = {-2,-1,-1,-1, 1, 1, 1, 2, 0, 0, 0}; // o[0] of each triple
__device__ __constant__ int kA2[11] = { 1, 0, 1, 2, 0,-1,-2,-1,-1, 0, 1}; // o[2] of each triple

__global__ __launch_bounds__(256) void shift_add(const float* __restrict__ temp,
                                                 float* __restrict__ out)
{
    const size_t idx = (size_t)blockIdx.x * 256 + threadIdx.x;   // 2^26 elements
    const int w = (int)(idx & 127);
    const int h = (int)((idx >> 7) & 127);
    const int t = (int)((idx >> 14) & 15);
    const int c = (int)((idx >> 18) & 127);
    const int b = (int)(idx >> 25);

    int h0 = 0, h2 = 0, w0 = 0, w2 = 0;
    if (c < 121) {
        const int i = c / 11;
        const int j = c - i * 11;
        h0 = kA0[i]; h2 = kA2[i];
        w0 = kA0[j]; w2 = kA2[j];
    }
    const int h1c = min(H_ - 1, max(0, h - h0));
    const int w1c = min(W_ - 1, max(0, w - w0));
    const int h3c = min(H_ - 1, max(0, h - h2));
    const int w3c = min(W_ - 1, max(0, w - w2));

    const size_t base = ((size_t)(b * COUT + c) * T_) * (H_ * W_);
    const float f2 = temp[base + (size_t)(t + 1) * (H_ * W_) + (size_t)h   * W_ + w  ];
    const float f1 = temp[base + (size_t)(t    ) * (H_ * W_) + (size_t)h1c * W_ + w1c];
    const float f3 = temp[base + (size_t)(t + 2) * (H_ * W_) + (size_t)h3c * W_ + w3c];

    out[idx] = f1 + f2 + f3;
}

// -------------------- launch --------------------
extern "C" void kernel_launch(void* const* d_in, const int* in_sizes, int n_in,
                              void* d_out, int out_size, void* d_ws, size_t ws_size,
                              hipStream_t stream)
{
    const float* x    = (const float*)d_in[0];
    const float* wgt  = (const float*)d_in[1];
    const float* bias = (const float*)d_in[2];
    float* temp = (float*)d_ws;          // needs 2*128*18*128*128*4 = 302 MB scratch
    float* out  = (float*)d_out;

    dim3 g1(W_ / NT, H_, B_ * T_);       // (2, 128, 36)
    conv3x3_wmma_f32<<<g1, 256, 0, stream>>>(x, wgt, bias, temp);

    const size_t nout = (size_t)B_ * COUT * 16 * H_ * W_;   // 2^26
    shift_add<<<(unsigned)(nout / 256), 256, 0, stream>>>(temp, out);
}